// DyRep_2345052144322
// MI455X (gfx1250) — compile-verified
//
#include <hip/hip_runtime.h>
#include <hip/hip_bf16.h>
#include <math.h>

typedef __attribute__((ext_vector_type(2))) float v2f;
typedef __attribute__((ext_vector_type(8))) float v8f;

#define N_NODES   50000
#define HDIM      32
#define B_EVENTS  16384
#define S_SAMP    20
#define CLAMPV    75.0f
#define N_TILES   (N_NODES / 16)   // 3125 exactly, no tail

// ---------------------------------------------------------------------------
// Kernel A: nd[n][k] = ws_k . emb[n]   (ws_k = 0.5*(W[k,:32]+W[k,32:]))
// A real (50000 x 32) @ (32 x 2) GEMM -> fp32 WMMA, one 16-row tile per wave,
// 8 chained V_WMMA_F32_16X16X4_F32 accumulate K=32. B operand (weights) is
// built branch-free and hoisted out of the chunk loop.
// ---------------------------------------------------------------------------
__global__ __launch_bounds__(256) void dyrep_node_dots_wmma(
    const float* __restrict__ emb,   // [N_NODES, 32]
    const float* __restrict__ W,     // [2, 64]
    float* __restrict__ nd)          // [N_NODES, 2]
{
  const int lane  = threadIdx.x & 31;
  const int wave  = threadIdx.x >> 5;
  const int tile  = blockIdx.x * 8 + wave;
  if (tile >= N_TILES) return;                 // uniform per wave -> EXEC stays full

  const int mn    = lane & 15;                 // A: M row index / B: N col index
  const int khalf = lane >> 4;                 // selects K pair {0,1} vs {2,3}
  const int rowBase = tile * 16;
  const long rowOff = (long)(rowBase + mn) * HDIM;

  // --- Hoisted B operand: lane&15 = N column; cols 0/1 = folded weights, rest 0.
  // Clamp the W row index so every lane loads in-bounds (W is 2x64), then
  // select 0 for N>=2 -> v_cndmask, no EXEC-mask dance inside the WMMA loop.
  const bool activeN = (mn < 2);
  const int  wrow    = mn & 1;
  v2f bop[8];
  #pragma unroll
  for (int ch = 0; ch < 8; ++ch) {
    const int k0 = ch * 4 + khalf * 2;
    const float2 wlo = *(const float2*)(W + wrow * 64 + k0);
    const float2 whi = *(const float2*)(W + wrow * 64 + 32 + k0);
    bop[ch].x = activeN ? 0.5f * (wlo.x + whi.x) : 0.f;
    bop[ch].y = activeN ? 0.5f * (wlo.y + whi.y) : 0.f;
  }

  // --- K=32 accumulation: 8 chained fp32 WMMAs, A chunk loaded as b64.
  v8f c = {0.f, 0.f, 0.f, 0.f, 0.f, 0.f, 0.f, 0.f};
  #pragma unroll
  for (int ch = 0; ch < 8; ++ch) {
    const int k0 = ch * 4 + khalf * 2;         // this lane's 2 K values in chunk
    // A 16x4 f32 layout: lanes 0-15 -> K = 4c+0,4c+1 ; lanes 16-31 -> 4c+2,4c+3
    const float2 av = *(const float2*)(emb + rowOff + k0);
    v2f a; a.x = av.x; a.y = av.y;
    c = __builtin_amdgcn_wmma_f32_16x16x4_f32(
        /*neg_a=*/false, a, /*neg_b=*/false, bop[ch],
        /*c_mod=*/(short)0, c, /*reuse_a=*/false, /*reuse_b=*/false);
  }

  // D 16x16 f32: VGPR r holds M = r + 8*khalf, lane&15 = N. Only N in {0,1} used.
  if (activeN) {
    #pragma unroll
    for (int r = 0; r < 8; ++r) {
      const int m = r + khalf * 8;
      nd[(rowBase + m) * 2 + mn] = c[r];
    }
  }
}

// ---------------------------------------------------------------------------
// Kernel B: per-event intensity + survival from precomputed per-node dots.
// ---------------------------------------------------------------------------
__device__ __forceinline__ float softplus_psi(float g, float psi, float inv_psi) {
  float r = g * inv_psi;
  r = fminf(fmaxf(r, -CLAMPV), CLAMPV);
  return psi * log1pf(__expf(r));
}

__global__ __launch_bounds__(256) void dyrep_events(
    const int* __restrict__ u, const int* __restrict__ v,
    const int* __restrict__ kk,
    const int* __restrict__ uo, const int* __restrict__ vo,
    const float* __restrict__ nd,      // [N_NODES,2]
    const float* __restrict__ b_omega, // [2]
    const float* __restrict__ psi,     // [2]
    float* __restrict__ out_int,       // [B]
    float* __restrict__ out_surv)      // [B]
{
  const int b = blockIdx.x * blockDim.x + threadIdx.x;
  if (b >= B_EVENTS) return;

  const float b0 = b_omega[0], b1 = b_omega[1];
  const float p0 = psi[0],     p1 = psi[1];
  const float ip0 = 1.0f / p0, ip1 = 1.0f / p1;

  const float2* nd2 = (const float2*)nd;
  const float2 ndu = nd2[u[b]];
  const float2 ndv = nd2[v[b]];

  // intensity at observed event type k[b]
  const int kb = kk[b];
  const float gk  = (kb == 0) ? (ndu.x + ndv.x + b0) : (ndu.y + ndv.y + b1);
  out_int[b] = softplus_psi(gk, (kb == 0) ? p0 : p1, (kb == 0) ? ip0 : ip1);

  // survival term over S negative samples, both event types
  const int* uoB = uo + b * S_SAMP;
  const int* voB = vo + b * S_SAMP;
  float acc = 0.f;
  #pragma unroll 4
  for (int s = 0; s < S_SAMP; ++s) {
    const float2 nvo = nd2[voB[s]];
    const float2 nuo = nd2[uoB[s]];
    acc += softplus_psi(ndu.x + nvo.x + b0, p0, ip0);
    acc += softplus_psi(ndu.y + nvo.y + b1, p1, ip1);
    acc += softplus_psi(ndv.x + nuo.x + b0, p0, ip0);
    acc += softplus_psi(ndv.y + nuo.y + b1, p1, ip1);
  }
  out_surv[b] = acc * (1.0f / (float)S_SAMP);
}

// ---------------------------------------------------------------------------
extern "C" void kernel_launch(void* const* d_in, const int* in_sizes, int n_in,
                              void* d_out, int out_size, void* d_ws, size_t ws_size,
                              hipStream_t stream) {
  (void)in_sizes; (void)n_in; (void)out_size; (void)ws_size;
  const int*   u    = (const int*)d_in[0];
  const int*   v    = (const int*)d_in[1];
  /* d_in[2] = t, unused by the reference outputs */
  const int*   kk   = (const int*)d_in[3];
  const int*   uo   = (const int*)d_in[4];   // [B,S]
  const int*   vo   = (const int*)d_in[5];   // [B,S]
  const float* emb  = (const float*)d_in[6]; // [N_NODES,32]
  const float* W    = (const float*)d_in[7]; // [2,64]
  const float* bom  = (const float*)d_in[8]; // [2]
  const float* psi  = (const float*)d_in[9]; // [2]

  float* out      = (float*)d_out;           // [0..B)=intensity, [B..2B)=survival
  float* nd       = (float*)d_ws;            // [N_NODES,2] scratch (400 KB)

  // Phase 1: per-node dot products via fp32 WMMA (8 waves/block, 1 tile/wave)
  const int blocksA = (N_TILES + 7) / 8;     // 391
  dyrep_node_dots_wmma<<<blocksA, 256, 0, stream>>>(emb, W, nd);

  // Phase 2: per-event softplus assembly
  const int blocksB = (B_EVENTS + 255) / 256; // 64
  dyrep_events<<<blocksB, 256, 0, stream>>>(u, v, kk, uo, vo, nd, bom, psi,
                                            out, out + B_EVENTS);
}